// GCNEncoder_17514876634164
// MI455X (gfx1250) — compile-verified
//
#include <hip/hip_runtime.h>
#include <hip/hip_bf16.h>

typedef __attribute__((ext_vector_type(2))) float v2f;
typedef __attribute__((ext_vector_type(8))) float v8f;

#define NNODES 100000

// ---------------------------------------------------------------------------
// Utility: fill a float buffer with a constant (grid-stride).
// ---------------------------------------------------------------------------
__global__ void fill_f32(float* __restrict__ p, float v, long long n) {
    long long i = (long long)blockIdx.x * blockDim.x + threadIdx.x;
    long long stride = (long long)gridDim.x * blockDim.x;
    for (; i < n; i += stride) p[i] = v;
}

// ---------------------------------------------------------------------------
// Degree accumulation: deg[dst[e]] += 1  (deg pre-filled with 1.0 = self loop)
// edge_index is [2, E] row-major: row 0 = src, row 1 = dst.
// ---------------------------------------------------------------------------
__global__ void deg_kernel(const int* __restrict__ ei, float* __restrict__ deg, int E) {
    int e = blockIdx.x * blockDim.x + threadIdx.x;
    if (e < E) {
        int d = ei[(long long)E + e];
        atomicAdd(&deg[d], 1.0f);
    }
}

// deg -> rsqrt(deg) in place (deg >= 1 always, so no zero guard needed)
__global__ void rsqrt_kernel(float* __restrict__ deg, int n) {
    int i = blockIdx.x * blockDim.x + threadIdx.x;
    if (i < n) deg[i] = rsqrtf(deg[i]);
}

// ---------------------------------------------------------------------------
// WMMA f32 GEMM: C[M,N] = A[M,K] * B[K,N], row-major, exact fp32 via
// V_WMMA_F32_16X16X4_F32. Each wave computes a 16(M) x 64(N) strip: one A
// fragment per K-step feeds 4 WMMAs into 4 independent accumulators, so A
// is fetched once per strip instead of once per 16x16 tile.
//
// A fragment (32-bit A 16x4, ISA layout): lanes 0-15 hold row M=lane with
// K = {k,k+1}; lanes 16-31 hold row M=lane-16 with K = {k+2,k+3}.
// B fragment (32-bit B 4x16, mirrored):   lanes 0-15 hold col N=lane with
// K = {k,k+1}; lanes 16-31 hold col N=lane-16 with K = {k+2,k+3}.
// C/D (8 VGPRs): VGPR r = row (tileM*16 + r + (lane>=16 ? 8 : 0)), col = lane&15.
// N must be a multiple of 64, M a multiple of 16. The out-of-range guard is
// wave-uniform so EXEC is all-1s at every WMMA (ISA requirement).
// ---------------------------------------------------------------------------
#define WMMA_F32(a, b, c) \
    __builtin_amdgcn_wmma_f32_16x16x4_f32(false, (a), false, (b), (short)0, (c), false, false)

__global__ void gemm_wmma_f32(const float* __restrict__ A,
                              const float* __restrict__ B,
                              float* __restrict__ C,
                              int M, int K, int N) {
    const int wavesPerBlock = blockDim.x >> 5;
    const int wave = blockIdx.x * wavesPerBlock + (threadIdx.x >> 5);
    const int lane = threadIdx.x & 31;
    const int stripsPerRow = N >> 6;                 // 64-wide N strips
    const int tileM = wave / stripsPerRow;
    const int strip = wave - tileM * stripsPerRow;
    if (tileM * 16 >= M) return;                     // uniform per wave

    const int lhalf = lane >> 4;                     // 0: K pair {0,1}, 1: {2,3}
    const int l15   = lane & 15;
    const int row   = tileM * 16 + l15;              // A row for this lane
    const int col0  = strip * 64 + l15;              // first B/C column

    const float* __restrict__ Arow = A + (long long)row * K;

    v8f acc0 = {}, acc1 = {}, acc2 = {}, acc3 = {};
    for (int k = 0; k < K; k += 4) {
        const int ka = k + lhalf * 2;
        v2f a;
        a.x = Arow[ka];
        a.y = Arow[ka + 1];

        const float* __restrict__ Br0 = B + (long long)ka * N;        // row ka
        const float* __restrict__ Br1 = Br0 + N;                      // row ka+1
        v2f b0, b1, b2, b3;
        b0.x = Br0[col0];      b0.y = Br1[col0];
        b1.x = Br0[col0 + 16]; b1.y = Br1[col0 + 16];
        b2.x = Br0[col0 + 32]; b2.y = Br1[col0 + 32];
        b3.x = Br0[col0 + 48]; b3.y = Br1[col0 + 48];

        acc0 = WMMA_F32(a, b0, acc0);
        acc1 = WMMA_F32(a, b1, acc1);
        acc2 = WMMA_F32(a, b2, acc2);
        acc3 = WMMA_F32(a, b3, acc3);
    }

    const int rbase = tileM * 16 + (lhalf ? 8 : 0);
#pragma unroll
    for (int r = 0; r < 8; ++r) {
        float* __restrict__ Crow = C + (long long)(rbase + r) * N;
        Crow[col0]      = acc0[r];
        Crow[col0 + 16] = acc1[r];
        Crow[col0 + 32] = acc2[r];
        Crow[col0 + 48] = acc3[r];
    }
}

// ---------------------------------------------------------------------------
// Edge scatter: one wave (32 lanes) per edge; lanes stride the C channels so
// row reads of H[src] are coalesced. out[dst,c] += H[src,c]*dinv[src]*dinv[dst]
// ---------------------------------------------------------------------------
__global__ void scatter_edges(const int* __restrict__ ei,
                              const float* __restrict__ H,
                              const float* __restrict__ dinv,
                              float* __restrict__ out,
                              int E, int C) {
    long long t = (long long)blockIdx.x * blockDim.x + threadIdx.x;
    long long e = t >> 5;
    int lane = threadIdx.x & 31;
    if (e >= E) return;
    int s = ei[e];
    int d = ei[(long long)E + e];
    float norm = dinv[s] * dinv[d];
    const float* __restrict__ hs = H + (long long)s * C;
    float* __restrict__ od = out + (long long)d * C;
    for (int c = lane; c < C; c += 32) {
        atomicAdd(&od[c], hs[c] * norm);
    }
}

// ---------------------------------------------------------------------------
// Self-loop + bias (+ optional ReLU), fused, fully coalesced:
//   agg[i,c] = act( agg[i,c] + H[i,c]*dinv[i]^2 + bias[c] )
// ---------------------------------------------------------------------------
__global__ void fixup_kernel(float* __restrict__ agg,
                             const float* __restrict__ H,
                             const float* __restrict__ dinv,
                             const float* __restrict__ bias,
                             long long total, int C, int doRelu) {
    long long i = (long long)blockIdx.x * blockDim.x + threadIdx.x;
    if (i >= total) return;
    int node = (int)(i / C);
    int c = (int)(i - (long long)node * C);
    float di = dinv[node];
    float v = agg[i] + H[i] * di * di + bias[c];
    if (doRelu) v = fmaxf(v, 0.0f);
    agg[i] = v;
}

// ---------------------------------------------------------------------------
// Host-side launcher
// ---------------------------------------------------------------------------
extern "C" void kernel_launch(void* const* d_in, const int* in_sizes, int n_in,
                              void* d_out, int out_size, void* d_ws, size_t ws_size,
                              hipStream_t stream) {
    const float* x   = (const float*)d_in[0];   // [100000,128]
    const int*   ei  = (const int*)d_in[1];     // [2, E]
    const float* W1  = (const float*)d_in[2];   // [128,128]
    const float* b1  = (const float*)d_in[3];   // [128]
    const float* W2  = (const float*)d_in[4];   // [128,64]
    const float* b2  = (const float*)d_in[5];   // [64]
    float* out = (float*)d_out;                 // [100000,64]

    const int E = in_sizes[1] / 2;
    const int M = NNODES;
    const int IN = 128, HID = 128, OUT = 64;

    // Workspace layout (floats):
    //   dinv : [0, 100000)         (padded region to 102400)
    //   H    : [102400, +12.8M)    gemm output, reused for layer-2 gemm
    //   AGG  : [+12.8M, +12.8M)    layer-1 aggregation buffer
    float* ws   = (float*)d_ws;
    float* dinv = ws;
    float* H    = ws + 102400;
    float* AGG  = H + (long long)M * HID;

    const long long nH1  = (long long)M * HID;   // 12.8M
    const long long nOut = (long long)M * OUT;   // 6.4M

    // ---- degree / normalization ----
    fill_f32<<<1024, 256, 0, stream>>>(dinv, 1.0f, (long long)M);      // self-loops
    deg_kernel<<<(E + 255) / 256, 256, 0, stream>>>(ei, dinv, E);
    rsqrt_kernel<<<(M + 255) / 256, 256, 0, stream>>>(dinv, M);

    // ---- layer 1: H = x @ W1 ----
    {
        int waves = (M / 16) * (HID / 64);            // 6250*2 = 12500 strips
        int blocks = (waves + 3) / 4;                 // 4 waves per 128-thr block
        gemm_wmma_f32<<<blocks, 128, 0, stream>>>(x, W1, H, M, IN, HID);
    }
    fill_f32<<<4096, 256, 0, stream>>>(AGG, 0.0f, nH1);
    {
        long long threads = (long long)E * 32;
        scatter_edges<<<(int)((threads + 255) / 256), 256, 0, stream>>>(ei, H, dinv, AGG, E, HID);
    }
    fixup_kernel<<<(int)((nH1 + 255) / 256), 256, 0, stream>>>(AGG, H, dinv, b1, nH1, HID, 1);

    // ---- layer 2: H2 = AGG @ W2 (H2 aliases H; H already consumed) ----
    {
        int waves = (M / 16) * (OUT / 64);            // 6250 strips
        int blocks = (waves + 3) / 4;
        gemm_wmma_f32<<<blocks, 128, 0, stream>>>(AGG, W2, H, M, HID, OUT);
    }
    fill_f32<<<4096, 256, 0, stream>>>(out, 0.0f, nOut);
    {
        long long threads = (long long)E * 32;
        scatter_edges<<<(int)((threads + 255) / 256), 256, 0, stream>>>(ei, H, dinv, out, E, OUT);
    }
    fixup_kernel<<<(int)((nOut + 255) / 256), 256, 0, stream>>>(out, H, dinv, b2, nOut, OUT, 0);
}